// attn_head_sparse_21165598834966
// MI455X (gfx1250) — compile-verified
//
#include <hip/hip_runtime.h>

#define N_NODES 50000
#define N_EDGES 1600000
#define FDIM 256
#define HDIM 256
#define LRELU_ALPHA 0.2f

typedef float v2f __attribute__((ext_vector_type(2)));
typedef float v8f __attribute__((ext_vector_type(8)));

// ---- order-preserving float <-> uint encoding for integer atomicMax ----
__device__ __forceinline__ unsigned fenc(float f) {
    unsigned u = __float_as_uint(f);
    return (u & 0x80000000u) ? ~u : (u | 0x80000000u);
}
__device__ __forceinline__ float fdec(unsigned u) {
    return (u & 0x80000000u) ? __uint_as_float(u ^ 0x80000000u)
                             : __uint_as_float(~u);
}
// enc(-inf) = ~0xFF800000 = 0x007FFFFF
#define MENC_NEGINF 0x007FFFFFu

// ---------------- kernel 0: init out=0, s=0, m=enc(-inf) ----------------
__global__ void k_init(float* __restrict__ out, float* __restrict__ s,
                       unsigned* __restrict__ menc) {
    int i = blockIdx.x * blockDim.x + threadIdx.x;
    if (i < N_NODES * HDIM) out[i] = 0.0f;
    if (i < N_NODES) { s[i] = 0.0f; menc[i] = MENC_NEGINF; }
}

// ---------------- kernel 1: h = x @ W via V_WMMA_F32_16X16X4_F32 ----------------
// One wave computes one 16x16 tile of h. 3125 row-tiles x 16 col-tiles = 50000 waves.
__global__ void k_gemm_wmma(const float* __restrict__ x,   // [N, F]
                            const float* __restrict__ W,   // [F, H]
                            float* __restrict__ h) {       // [N, H]
    int wid  = (blockIdx.x * blockDim.x + threadIdx.x) >> 5;  // global wave id
    if (wid >= (N_NODES / 16) * (HDIM / 16)) return;
    int lane = threadIdx.x & 31;
    int half = lane >> 4;      // 0: lanes 0-15, 1: lanes 16-31
    int l    = lane & 15;

    int rowTile = wid >> 4;            // /16 col tiles
    int colTile = wid & 15;
    int row0 = rowTile * 16;
    int col0 = colTile * 16;

    const float* Arow = x + (size_t)(row0 + l) * FDIM;   // A: M = l
    const float* Bcol = W + col0 + l;                    // B: N = l

    v8f acc = {};
    #pragma unroll 4
    for (int k = 0; k < FDIM; k += 4) {
        // A 16x4 f32: VGPR0 = A[m][k + 2*half], VGPR1 = A[m][k + 2*half + 1]
        v2f a;
        a.x = Arow[k + 2 * half];
        a.y = Arow[k + 2 * half + 1];
        // B 4x16 f32: VGPR0 = B[k + 2*half][n], VGPR1 = B[k + 2*half + 1][n]
        v2f b;
        b.x = Bcol[(size_t)(k + 2 * half) * HDIM];
        b.y = Bcol[(size_t)(k + 2 * half + 1) * HDIM];
        // (neg_a, A, neg_b, B, c_mod, C, reuse_a, reuse_b)
        acc = __builtin_amdgcn_wmma_f32_16x16x4_f32(
            false, a, false, b, (short)0, acc, false, false);
    }

    // C/D 16x16 f32: VGPR j holds row (j + 8*half), col = l
    int rbase = row0 + half * 8;
    #pragma unroll
    for (int j = 0; j < 8; ++j)
        h[(size_t)(rbase + j) * HDIM + col0 + l] = acc[j];
}

// ---------------- kernel 2: f1 = h@a1 + b1, f2 = h@a2 + b2 (one wave / node) ----
__global__ void k_fdot(const float* __restrict__ h,
                       const float* __restrict__ a1, const float* __restrict__ b1,
                       const float* __restrict__ a2, const float* __restrict__ b2,
                       float* __restrict__ f1, float* __restrict__ f2) {
    int node = (blockIdx.x * blockDim.x + threadIdx.x) >> 5;
    int lane = threadIdx.x & 31;
    if (node >= N_NODES) return;
    const float* hr = h + (size_t)node * HDIM;
    float s1 = 0.0f, s2 = 0.0f;
    #pragma unroll
    for (int j = lane; j < HDIM; j += 32) {
        float v = hr[j];
        s1 += v * a1[j];
        s2 += v * a2[j];
    }
    #pragma unroll
    for (int off = 16; off > 0; off >>= 1) {
        s1 += __shfl_xor(s1, off, 32);
        s2 += __shfl_xor(s2, off, 32);
    }
    if (lane == 0) {
        f1[node] = s1 + b1[0];
        f2[node] = s2 + b2[0];
    }
}

// ---------------- kernel 3: e = lrelu(ev*(f1[r]+f2[c])), segment max -------------
__global__ void k_edge_lrelu_max(const int* __restrict__ ei,
                                 const float* __restrict__ ev,
                                 const float* __restrict__ f1,
                                 const float* __restrict__ f2,
                                 float* __restrict__ ebuf,
                                 unsigned* __restrict__ menc) {
    int e = blockIdx.x * blockDim.x + threadIdx.x;
    if (e >= N_EDGES) return;
    int r = ei[2 * e];
    int c = ei[2 * e + 1];
    float v = ev[e] * (f1[r] + f2[c]);
    v = v > 0.0f ? v : LRELU_ALPHA * v;
    ebuf[e] = v;
    atomicMax(&menc[r], fenc(v));
}

// ---------------- kernel 4: ex = exp(e - m[r]); s[r] += ex ----------------------
__global__ void k_edge_expsum(const int* __restrict__ ei,
                              const float* __restrict__ ebuf,
                              const unsigned* __restrict__ menc,
                              float* __restrict__ s) {
    int e = blockIdx.x * blockDim.x + threadIdx.x;
    if (e >= N_EDGES) return;
    int r = ei[2 * e];
    float ex = expf(ebuf[e] - fdec(menc[r]));
    atomicAdd(&s[r], ex);
}

// ---------------- kernel 5: out[r] += (ex/s[r]) * h[c]  (one wave / edge) -------
__global__ void k_edge_aggregate(const int* __restrict__ ei,
                                 const float* __restrict__ ebuf,
                                 const unsigned* __restrict__ menc,
                                 const float* __restrict__ s,
                                 const float* __restrict__ h,
                                 float* __restrict__ out) {
    int e    = (blockIdx.x * blockDim.x + threadIdx.x) >> 5;
    int lane = threadIdx.x & 31;
    if (e >= N_EDGES) return;
    int r = ei[2 * e];
    int c = ei[2 * e + 1];
    float ex = expf(ebuf[e] - fdec(menc[r]));
    float sv = s[r];
    float coef = ex / (sv == 0.0f ? 1.0f : sv);
    const float* hc = h + (size_t)c * HDIM;
    float* orow = out + (size_t)r * HDIM;
    #pragma unroll
    for (int j = lane; j < HDIM; j += 32)
        atomicAdd(&orow[j], coef * hc[j]);
}

// ---------------- kernel 6: ELU in place ----------------------------------------
__global__ void k_elu(float* __restrict__ out) {
    int i = blockIdx.x * blockDim.x + threadIdx.x;
    if (i >= N_NODES * HDIM) return;
    float v = out[i];
    out[i] = v > 0.0f ? v : expm1f(v);
}

extern "C" void kernel_launch(void* const* d_in, const int* in_sizes, int n_in,
                              void* d_out, int out_size, void* d_ws, size_t ws_size,
                              hipStream_t stream) {
    const float* x   = (const float*)d_in[0];   // seq (1,N,F)
    const int*   ei  = (const int*)  d_in[1];   // edge_index (E,2)
    const float* ev  = (const float*)d_in[2];   // edge_vals (E)
    const float* W   = (const float*)d_in[3];   // (F,H)
    const float* a1  = (const float*)d_in[4];
    const float* b1  = (const float*)d_in[5];
    const float* a2  = (const float*)d_in[6];
    const float* b2  = (const float*)d_in[7];
    // d_in[8] = bias_zero (all zeros) -> no-op

    // workspace layout (floats): h[N*H] | f1[N] | f2[N] | menc[N] | s[N] | ebuf[E]
    float*    h    = (float*)d_ws;
    float*    f1   = h + (size_t)N_NODES * HDIM;
    float*    f2   = f1 + N_NODES;
    unsigned* menc = (unsigned*)(f2 + N_NODES);
    float*    s    = (float*)(menc + N_NODES);
    float*    ebuf = s + N_NODES;

    float* out = (float*)d_out;

    const int B = 256;
    int nh_blocks   = (N_NODES * HDIM + B - 1) / B;          // 50000
    int gemm_blocks = ((N_NODES / 16) * (HDIM / 16) * 32 + B - 1) / B; // 6250
    int node_blocks = (N_NODES * 32 + B - 1) / B;            // 6250
    int edge_blocks = (N_EDGES + B - 1) / B;                 // 6250
    int agg_blocks  = (N_EDGES * 32 + B - 1) / B;            // 200000 (int ok: 5.12e7 threads)

    k_init<<<nh_blocks, B, 0, stream>>>(out, s, menc);
    k_gemm_wmma<<<gemm_blocks, B, 0, stream>>>(x, W, h);
    k_fdot<<<node_blocks, B, 0, stream>>>(h, a1, b1, a2, b2, f1, f2);
    k_edge_lrelu_max<<<edge_blocks, B, 0, stream>>>(ei, ev, f1, f2, ebuf, menc);
    k_edge_expsum<<<edge_blocks, B, 0, stream>>>(ei, ebuf, menc, s);
    k_edge_aggregate<<<agg_blocks, B, 0, stream>>>(ei, ebuf, menc, s, h, out);
    k_elu<<<nh_blocks, B, 0, stream>>>(out);
}